// BKTNN_75582834475152
// MI455X (gfx1250) — compile-verified
//
#include <hip/hip_runtime.h>
#include <hip/hip_bf16.h>
#include <math.h>

// Problem constants (fixed by the reference).
constexpr int   kB   = 8192;
constexpr int   kT   = 1024;
constexpr int   kH   = 64;
constexpr float kEps = 1e-6f;
constexpr int   kCH  = 32;     // time-chunk per LDS staging pass

typedef float v2f __attribute__((ext_vector_type(2)));
typedef float v8f __attribute__((ext_vector_type(8)));

// ---------------------------------------------------------------------------
// Kernel 1: embedding gather + 2x (64x64 GEMM + bias + ReLU) via f32 WMMA,
//           then 64x4 head + sigmoid + clip -> params[B][4] = (l,g,s,prior).
// grid = B/128 blocks x 256 threads; each wave owns a 16-row tile and
// computes a 16x64 output slab as four 16x16 WMMA accumulators.
// K-outer / N-inner loop: each A fragment is loaded from LDS once and fed to
// 4 independent WMMAs (independent accumulators -> no RAW stalls on the XDL).
// ---------------------------------------------------------------------------
__global__ __launch_bounds__(256)
void bkt_mlp_params(const int* __restrict__ X,
                    const float* __restrict__ embed,
                    const float* __restrict__ W0, const float* __restrict__ b0,
                    const float* __restrict__ W1, const float* __restrict__ b1,
                    const float* __restrict__ Wout, const float* __restrict__ bout,
                    float* __restrict__ params)
{
    __shared__ float sW[kH][kH + 1];    // current layer weights, [k][n]
    __shared__ float sH[128][kH + 1];   // activations for this block's 128 rows
    __shared__ float sWo[kH * 4];       // Wout
    __shared__ float sBias[kH];
    __shared__ int   sX[128];

    const int tid  = threadIdx.x;
    const int wave = tid >> 5;
    const int lane = tid & 31;
    const int r0   = blockIdx.x * 128;

    if (tid < 128)     sX[tid]  = X[r0 + tid];
    if (tid < kH * 4)  sWo[tid] = Wout[tid];
    __syncthreads();

    // Gather embedding rows (coalesced: each wave reads 128B runs).
    for (int i = 0; i < 32; ++i) {
        int flat = i * 256 + tid;
        int row  = flat >> 6, col = flat & 63;
        sH[row][col] = embed[(size_t)sX[row] * kH + col];
    }

    // WMMA fragment coordinates (ISA layouts):
    //   A 16x4 f32 : lane -> M = lane&15,           K = 2*(lane>>4) + r
    //   B 4x16 f32 : lane -> K = 2*(lane>>4) + r,   N = lane&15
    //   C 16x16 f32: lane -> N = lane&15,           M = r + 8*(lane>>4)
    const int m     = (wave << 4) + (lane & 15);
    const int khalf = (lane >> 4) << 1;            // 0 or 2
    const int ncol  = lane & 15;

    for (int layer = 0; layer < 2; ++layer) {
        const float* W  = layer ? W1 : W0;
        const float* bb = layer ? b1 : b0;
        for (int i = 0; i < 16; ++i) {            // 64x64 weights -> LDS
            int flat = i * 256 + tid;
            sW[flat >> 6][flat & 63] = W[flat];
        }
        if (tid < kH) sBias[tid] = bb[tid];
        __syncthreads();

        v8f acc0 = {0.f,0.f,0.f,0.f,0.f,0.f,0.f,0.f};
        v8f acc1 = acc0, acc2 = acc0, acc3 = acc0;

#if __has_builtin(__builtin_amdgcn_wmma_f32_16x16x4_f32)
#pragma unroll 4
        for (int kk = 0; kk < 16; ++kk) {
            const int kb = kk * 4 + khalf;
            const v2f a  = { sH[m][kb], sH[m][kb + 1] };
            const v2f b0f = { sW[kb][ 0 + ncol], sW[kb + 1][ 0 + ncol] };
            const v2f b1f = { sW[kb][16 + ncol], sW[kb + 1][16 + ncol] };
            const v2f b2f = { sW[kb][32 + ncol], sW[kb + 1][32 + ncol] };
            const v2f b3f = { sW[kb][48 + ncol], sW[kb + 1][48 + ncol] };
            acc0 = __builtin_amdgcn_wmma_f32_16x16x4_f32(false, a, false, b0f,
                                                         (short)0, acc0, false, false);
            acc1 = __builtin_amdgcn_wmma_f32_16x16x4_f32(false, a, false, b1f,
                                                         (short)0, acc1, false, false);
            acc2 = __builtin_amdgcn_wmma_f32_16x16x4_f32(false, a, false, b2f,
                                                         (short)0, acc2, false, false);
            acc3 = __builtin_amdgcn_wmma_f32_16x16x4_f32(false, a, false, b3f,
                                                         (short)0, acc3, false, false);
        }
#else
        // Scalar fallback producing the same C-fragment layout.
        for (int r = 0; r < 8; ++r) {
            const int row = (wave << 4) + r + ((lane >> 4) << 3);
            float s0 = 0.f, s1 = 0.f, s2 = 0.f, s3 = 0.f;
            for (int k = 0; k < kH; ++k) {
                const float h = sH[row][k];
                s0 = fmaf(h, sW[k][ 0 + ncol], s0);
                s1 = fmaf(h, sW[k][16 + ncol], s1);
                s2 = fmaf(h, sW[k][32 + ncol], s2);
                s3 = fmaf(h, sW[k][48 + ncol], s3);
            }
            acc0[r] = s0; acc1[r] = s1; acc2[r] = s2; acc3[r] = s3;
        }
#endif
        __syncthreads();   // all waves done reading sW / sH for this layer

        // bias + ReLU, write activations back (C layout: M = r + 8*(lane>=16))
        {
            const float bias0 = sBias[ 0 + ncol];
            const float bias1 = sBias[16 + ncol];
            const float bias2 = sBias[32 + ncol];
            const float bias3 = sBias[48 + ncol];
#pragma unroll
            for (int r = 0; r < 8; ++r) {
                const int row = (wave << 4) + r + ((lane >> 4) << 3);
                const float v0 = acc0[r] + bias0;
                const float v1 = acc1[r] + bias1;
                const float v2 = acc2[r] + bias2;
                const float v3 = acc3[r] + bias3;
                sH[row][ 0 + ncol] = v0 > 0.f ? v0 : 0.f;
                sH[row][16 + ncol] = v1 > 0.f ? v1 : 0.f;
                sH[row][32 + ncol] = v2 > 0.f ? v2 : 0.f;
                sH[row][48 + ncol] = v3 > 0.f ? v3 : 0.f;
            }
        }
        __syncthreads();
    }

    // Output head: 128 rows x 4 cols = 512 dot products of length 64.
    for (int rep = 0; rep < 2; ++rep) {
        const int q = rep * 256 + tid;
        const int row = q >> 2, col = q & 3;
        float sum = bout[col];
#pragma unroll 8
        for (int k = 0; k < kH; ++k)
            sum = fmaf(sH[row][k], sWo[k * 4 + col], sum);
        float sg = 1.0f / (1.0f + __expf(-sum));
        sg = fminf(fmaxf(sg, kEps), 1.0f - kEps);
        params[(size_t)(r0 + row) * 4 + col] = sg;
    }
}

// ---------------------------------------------------------------------------
// Kernel 2: sequential BKT scan, one thread per row, LDS-staged coalesced I/O.
// grid = B/256 blocks x 256 threads. Row pitch 36 keeps int4/float4 tiles
// 16B-aligned and limits compute-phase bank conflicts to 2-way.
// ---------------------------------------------------------------------------
__global__ __launch_bounds__(256)
void bkt_scan(const int* __restrict__ y,
              const float* __restrict__ params,
              float* __restrict__ corrects,
              float* __restrict__ latents)
{
    __shared__ __align__(16) int   sY[256][36];
    __shared__ __align__(16) float sC[256][36];
    __shared__ __align__(16) float sL[256][36];

    const int tid = threadIdx.x;
    const int r0  = blockIdx.x * 256;
    const int row = r0 + tid;

    const float4 p = *(const float4*)(params + (size_t)row * 4);
    const float l = p.x, g = p.y, s = p.z;
    float latent  = p.w;                     // prior (already clipped)
    const float oms = 1.0f - s;
    const float omg = 1.0f - g;
    const float oml = 1.0f - l;
    const float A1  = oms - g;               // 1 - s - g

    for (int t0 = 0; t0 < kT; t0 += kCH) {
        // Cooperative coalesced load: 256 rows x 32 steps, 128-bit per lane.
#pragma unroll
        for (int i = 0; i < 8; ++i) {
            int flat = i * 256 + tid;        // int4 index, 2048 total
            int rr = flat >> 3, c4 = flat & 7;
            int4 v = *(const int4*)(y + (size_t)(r0 + rr) * kT + t0 + c4 * 4);
            *(int4*)&sY[rr][c4 * 4] = v;
        }
        __syncthreads();

        if (t0 + kCH < kT)                   // warm next chunk (global_prefetch_b8)
            __builtin_prefetch(y + (size_t)row * kT + t0 + kCH, 0, 0);

#pragma unroll 4
        for (int i = 0; i < kCH; ++i) {
            int   yt      = sY[tid][i];      // -1/0/1; true_correct>0.5 <=> yt==1
            float correct = fmaf(latent, A1, g);
            bool  one     = yt > 0;
            float num     = latent * (one ? oms : s);
            float den     = one ? correct : fmaf(-latent, A1, omg);
            float k       = num / den;
            sC[tid][i] = correct;
            sL[tid][i] = latent;             // pre-update latent is the output
            latent = fminf(fmaxf(fmaf(k, oml, l), kEps), 1.0f - kEps);
        }
        __syncthreads();

        // Cooperative coalesced stores of both outputs.
#pragma unroll
        for (int i = 0; i < 8; ++i) {
            int flat = i * 256 + tid;
            int rr = flat >> 3, c4 = flat & 7;
            size_t go = (size_t)(r0 + rr) * kT + t0 + c4 * 4;
            *(float4*)(corrects + go) = *(const float4*)&sC[rr][c4 * 4];
            *(float4*)(latents  + go) = *(const float4*)&sL[rr][c4 * 4];
        }
        __syncthreads();
    }
}

// ---------------------------------------------------------------------------
extern "C" void kernel_launch(void* const* d_in, const int* in_sizes, int n_in,
                              void* d_out, int out_size, void* d_ws, size_t ws_size,
                              hipStream_t stream)
{
    const int*   X     = (const int*)d_in[0];
    const int*   y     = (const int*)d_in[1];
    const float* embed = (const float*)d_in[2];
    const float* W0    = (const float*)d_in[3];
    const float* b0    = (const float*)d_in[4];
    const float* W1    = (const float*)d_in[5];
    const float* b1    = (const float*)d_in[6];
    const float* Wout  = (const float*)d_in[7];
    const float* bout  = (const float*)d_in[8];

    float* params   = (float*)d_ws;                  // B*4 floats = 128 KB
    float* corrects = (float*)d_out;                 // outputs concatenated:
    float* latents  = corrects + (size_t)kB * kT;    // corrects.T then latents.T

    bkt_mlp_params<<<kB / 128, 256, 0, stream>>>(X, embed, W0, b0, W1, b1,
                                                 Wout, bout, params);
    bkt_scan<<<kB / 256, 256, 0, stream>>>(y, params, corrects, latents);
}